// Routerv3_85401129714221
// MI455X (gfx1250) — compile-verified
//
#include <hip/hip_runtime.h>
#include <hip/hip_bf16.h>

typedef __attribute__((ext_vector_type(2))) float v2f;
typedef __attribute__((ext_vector_type(8))) float v8f;

#define BSZ 64
#define LSZ 512
#define DSZ 768
#define HSZ 12
#define KSZ 256
#define TSZ (KSZ + 2)   // 258 output tokens

// ---------------------------------------------------------------------------
// Kernel 1: scores[b,j] = (1/H) * sum_{h,i} attn[b,h,i,j]
// WMMA trick: D = ones(16x4) @ B(4x16) + C  => every row of D is colsum(B).
// Each wave handles 16 columns; block = 256 threads = 8 waves = 128 columns.
// grid.x = B * (L/128). Column-sum is permutation-invariant in K, so the
// exact K<->lane-half mapping of the B operand cannot change the result.
// ---------------------------------------------------------------------------
__global__ void scores_kernel(const float* __restrict__ attn,
                              float* __restrict__ scores) {
    const int tid    = threadIdx.x;
    const int wave   = tid >> 5;
    const int lane   = tid & 31;
    const int tiles  = LSZ / 128;
    const int b      = blockIdx.x / tiles;
    const int jbase  = (blockIdx.x % tiles) * 128 + wave * 16;
    const int col    = jbase + (lane & 15);
    const int rowOff = (lane >> 4) * 2;          // lanes 0-15 rows {r,r+1}, lanes 16-31 {r+2,r+3}

    const float* base = attn + (size_t)b * HSZ * LSZ * LSZ + col;

    v2f a; a[0] = 1.0f; a[1] = 1.0f;             // ones A-matrix (16x4)
    v8f acc0 = {}; v8f acc1 = {};

    const float* p0 = base + (size_t)(rowOff)     * LSZ;
    const float* p1 = base + (size_t)(rowOff + 4) * LSZ;
    const size_t step = (size_t)8 * LSZ;

    // H*L = 6144 rows, 8 rows per iteration (two independent WMMA chains)
    for (int r = 0; r < HSZ * LSZ; r += 8) {
        v2f b0; b0[0] = p0[0]; b0[1] = p0[LSZ];
        v2f b1; b1[0] = p1[0]; b1[1] = p1[LSZ];
        acc0 = __builtin_amdgcn_wmma_f32_16x16x4_f32(false, a, false, b0,
                                                     (short)0, acc0, false, false);
        acc1 = __builtin_amdgcn_wmma_f32_16x16x4_f32(false, a, false, b1,
                                                     (short)0, acc1, false, false);
        p0 += step; p1 += step;
    }
    // D layout: VGPR0, lanes 0-15 hold (M=0, N=lane) -> colsum for column `col`
    if (lane < 16) {
        scores[b * LSZ + col] = (acc0[0] + acc1[0]) * (1.0f / (float)HSZ);
    }
}

// ---------------------------------------------------------------------------
// Kernel 2: stable top-K ranking per batch (one block of 512 threads / batch).
// impRank_j  = #{k: s_k > s_j} + #{k: s_k == s_j && k < j}  (matches lax.top_k)
// unimpRank  = mirrored for top_k(-scores).
// Prefix scan of "important" flags -> ascending-index output slot (== sorted
// imp_idx). Also emits the output attention mask and the unimportant count.
// ---------------------------------------------------------------------------
__global__ void topk_kernel(const float* __restrict__ scores,
                            const float* __restrict__ am,
                            int* __restrict__ imp_idx,
                            unsigned char* __restrict__ unimp_flag,
                            int* __restrict__ unimp_count,
                            float* __restrict__ out_mask) {
    __shared__ float s[LSZ];
    __shared__ int   scan[LSZ];
    __shared__ int   unCnt;

    const int b = blockIdx.x;
    const int j = threadIdx.x;

    float sj = scores[b * LSZ + j];
    s[j] = sj;
    if (j == 0) unCnt = 0;
    __syncthreads();

    int rImp = 0, rUn = 0;
    for (int k = 0; k < LSZ; ++k) {
        float sk = s[k];
        rImp += (sk > sj) || (sk == sj && k < j);
        rUn  += (sk < sj) || (sk == sj && k < j);
    }
    const int isImp = (rImp < KSZ) ? 1 : 0;
    const int isUn  = (rUn  < (LSZ - KSZ)) ? 1 : 0;

    unimp_flag[b * LSZ + j] = (unsigned char)isUn;
    if (isUn) atomicAdd(&unCnt, 1);

    scan[j] = isImp;
    __syncthreads();
    // Hillis-Steele inclusive scan
    for (int off = 1; off < LSZ; off <<= 1) {
        int v   = scan[j];
        int add = (j >= off) ? scan[j - off] : 0;
        __syncthreads();
        scan[j] = v + add;
        __syncthreads();
    }
    if (isImp) {
        int pos = scan[j] - 1;                     // exclusive prefix
        imp_idx[b * KSZ + pos] = j;
        out_mask[b * TSZ + 1 + pos] = am[b * LSZ + j];
    }
    if (j == 0) {
        out_mask[b * TSZ]           = 0.0f;        // class token slot
        out_mask[b * TSZ + KSZ + 1] = 0.0f;        // new token slot
        unimp_count[b] = 0;                        // placeholder, fixed below
    }
    __syncthreads();
    if (j == 0) unimp_count[b] = unCnt;
}

// ---------------------------------------------------------------------------
// Kernel 3: assemble output tokens. One 256-thread block per (b, t) row.
// t==0: class token; 1..K: gathered preserved rows; t==K+1: mean of
// unimportant rows (flags staged in LDS; flag is block-uniform per j so the
// branch is scalar and skips loads of kept rows).
// ---------------------------------------------------------------------------
__global__ void assemble_kernel(const float* __restrict__ hidden,
                                const int* __restrict__ imp_idx,
                                const unsigned char* __restrict__ unimp_flag,
                                const int* __restrict__ unimp_count,
                                float* __restrict__ out_tok) {
    const int b   = blockIdx.x / TSZ;
    const int t   = blockIdx.x % TSZ;
    const int tid = threadIdx.x;
    float* dst = out_tok + ((size_t)b * TSZ + t) * DSZ;

    if (t < TSZ - 1) {
        const int src = (t == 0) ? 0 : imp_idx[b * KSZ + (t - 1)];
        const float* srcp = hidden + ((size_t)b * LSZ + src) * DSZ;
        dst[tid]       = srcp[tid];
        dst[tid + 256] = srcp[tid + 256];
        dst[tid + 512] = srcp[tid + 512];
    } else {
        __shared__ unsigned char fl[LSZ];
        fl[tid]       = unimp_flag[b * LSZ + tid];
        fl[tid + 256] = unimp_flag[b * LSZ + tid + 256];
        __syncthreads();
        float a0 = 0.f, a1 = 0.f, a2 = 0.f;
        const float* hp = hidden + (size_t)b * LSZ * DSZ;
        for (int j = 0; j < LSZ; ++j) {
            if (fl[j]) {
                const float* row = hp + (size_t)j * DSZ;
                a0 += row[tid];
                a1 += row[tid + 256];
                a2 += row[tid + 512];
            }
        }
        const float inv = 1.0f / (float)unimp_count[b];
        dst[tid]       = a0 * inv;
        dst[tid + 256] = a1 * inv;
        dst[tid + 512] = a2 * inv;
    }
}

// ---------------------------------------------------------------------------
extern "C" void kernel_launch(void* const* d_in, const int* in_sizes, int n_in,
                              void* d_out, int out_size, void* d_ws, size_t ws_size,
                              hipStream_t stream) {
    const float* hidden = (const float*)d_in[0];
    const float* am     = (const float*)d_in[1];   // (B,1,1,L) == flat (B,L)
    const float* attn   = (const float*)d_in[2];
    // d_in[3] is K (==256, baked into KSZ)

    const int B = in_sizes[0] / (LSZ * DSZ);

    float* out_tok  = (float*)d_out;
    float* out_mask = out_tok + (size_t)B * TSZ * DSZ;

    char* ws = (char*)d_ws;
    float*         scores      = (float*)ws;
    int*           imp_idx     = (int*)(ws + (size_t)B * LSZ * 4);
    unsigned char* unimp_flag  = (unsigned char*)(ws + (size_t)B * LSZ * 4 + (size_t)B * KSZ * 4);
    int*           unimp_count = (int*)(ws + (size_t)B * LSZ * 4 + (size_t)B * KSZ * 4 + (size_t)B * LSZ);

    scores_kernel<<<dim3(B * (LSZ / 128)), dim3(256), 0, stream>>>(attn, scores);
    topk_kernel<<<dim3(B), dim3(LSZ), 0, stream>>>(scores, am, imp_idx, unimp_flag,
                                                   unimp_count, out_mask);
    assemble_kernel<<<dim3(B * TSZ), dim3(256), 0, stream>>>(hidden, imp_idx, unimp_flag,
                                                             unimp_count, out_tok);
}